// QuantumSelfAttention_65481071395826
// MI455X (gfx1250) — compile-verified
//
#include <hip/hip_runtime.h>
#include <hip/hip_bf16.h>

typedef float v2f __attribute__((ext_vector_type(2)));
typedef float v8f __attribute__((ext_vector_type(8)));

#define CA 0.92387953251128675613f /* cos(pi/8) */
#define SA 0.38268343236508977173f /* sin(pi/8) */

__global__ __launch_bounds__(256) void qsa_wmma_kernel(const float* __restrict__ features,
                                                       float* __restrict__ out,
                                                       int rows) {
    __shared__ float Ur[16][16];
    __shared__ float Ui[16][16];
    __shared__ float Amat[16][17];   // +1 pad to dodge bank conflicts

    const int tid = threadIdx.x;

    // --- Build columns of the fixed CRX-ladder unitary U (threads 0..15) ---
    if (tid < 16) {
        float sr[16], si[16];
#pragma unroll
        for (int i = 0; i < 16; ++i) { sr[i] = (i == tid) ? 1.0f : 0.0f; si[i] = 0.0f; }
#pragma unroll
        for (int g = 0; g < 3; ++g) {            // CRX(q=g -> q=g+1)
            const int cb = 3 - g;                // control bit position
            const int tb = 2 - g;                // target bit position
#pragma unroll
            for (int i = 0; i < 16; ++i) {
                if (((i >> cb) & 1) && !((i >> tb) & 1)) {
                    const int j = i | (1 << tb);
                    const float s0r = sr[i], s0i = si[i], s1r = sr[j], s1i = si[j];
                    sr[i] = CA * s0r + SA * s1i;   // RX(pi/4) on (i,j) pair
                    si[i] = CA * s0i - SA * s1r;
                    sr[j] = CA * s1r + SA * s0i;
                    si[j] = CA * s1i - SA * s0r;
                }
            }
        }
#pragma unroll
        for (int i = 0; i < 16; ++i) { Ur[i][tid] = sr[i]; Ui[i][tid] = si[i]; }
    }
    __syncthreads();

    // --- A[m][k] = sum_{i=8..15} Ur[i][m]Ur[i][k] + Ui[i][m]Ui[i][k]  (symmetric) ---
    {
        const int m = tid >> 4, k = tid & 15;
        float acc = 0.0f;
#pragma unroll
        for (int i = 8; i < 16; ++i)
            acc += Ur[i][m] * Ur[i][k] + Ui[i][m] * Ui[i][k];
        Amat[m][k] = acc;
    }
    __syncthreads();

    const int wave = tid >> 5;
    const int lane = tid & 31;
    const int n    = lane & 15;     // circuit column / A-matrix row owned by this lane
    const int half = lane >> 4;     // 0: K=0,1 / M=0..7 ; 1: K=2,3 / M=8..15

    // A-operand (loop-invariant): lane holds A[n][4kb + 2*half .. +1]
    v2f aop[4];
#pragma unroll
    for (int kb = 0; kb < 4; ++kb) {
        aop[kb][0] = Amat[n][4 * kb + 2 * half + 0];
        aop[kb][1] = Amat[n][4 * kb + 2 * half + 1];
    }

    const int waveStride = gridDim.x * 8;         // waves in grid
    for (int tile = blockIdx.x * 8 + wave; tile * 16 < rows; tile += waveStride) {
        const int row  = tile * 16 + n;
        const int rowc = row < rows ? row : rows - 1;        // keep EXEC full
        const float4 ang = *(const float4*)(features + (size_t)rowc * 64);

        float c0, s0, c1, s1, c2, s2, c3, s3;
        sincosf(0.5f * ang.x, &s0, &c0);
        sincosf(0.5f * ang.y, &s1, &c1);
        sincosf(0.5f * ang.z, &s2, &c2);
        sincosf(0.5f * ang.w, &s3, &c3);

        // pab[a*2+b] = f0[a]*f1[b], pcd[c*2+d] = f2[c]*f3[d]
        const float pab[4] = { c0 * c1, c0 * s1, s0 * c1, s0 * s1 };
        const float pcd[4] = { c2 * c3, c2 * s3, s2 * c3, s2 * s3 };

        // Half-dependent factors selected ONCE (no dynamic psi[] indexing):
        // B-op entry i = 4kb+2*half+d  ->  pab[kb] * f2[half] * f3[d]
        const float c2h = half ? s2 : c2;
        const float f0h = half ? s0 : c0;
        const float t0 = c2h * c3;
        const float t1 = c2h * s3;

        // Y = A * Psi : 4 chained fp32 WMMAs (16x16x4 each)
        v8f acc = {};
#pragma unroll
        for (int kb = 0; kb < 4; ++kb) {
            v2f bop;
            bop[0] = pab[kb] * t0;
            bop[1] = pab[kb] * t1;
            acc = __builtin_amdgcn_wmma_f32_16x16x4_f32(
                /*neg_a=*/false, aop[kb], /*neg_b=*/false, bop,
                /*c_mod=*/(short)0, acc, /*reuse_a=*/false, /*reuse_b=*/false);
        }

        // p1 = psi^T Y ; lane holds Y[r+8*half][n] in acc[r],
        // psi[r+8*half] = f0h * f1[r>>2] * pcd[r&3]  (all constant indices)
        const float f1v[2] = { c1, s1 };
        float dotv = 0.0f;
#pragma unroll
        for (int r = 0; r < 8; ++r)
            dotv += (f1v[r >> 2] * pcd[r & 3]) * acc[r];
        const float partial = f0h * dotv;

        const float p1 = partial + __shfl_xor(partial, 16, 32);
        if (half == 0 && row < rows) out[row] = p1;
    }
}

extern "C" void kernel_launch(void* const* d_in, const int* in_sizes, int n_in,
                              void* d_out, int out_size, void* d_ws, size_t ws_size,
                              hipStream_t stream) {
    const float* features = (const float*)d_in[0];
    float* out = (float*)d_out;
    const int rows  = in_sizes[0] / 64;            // B*N  (D = 64)
    const int tiles = (rows + 15) / 16;
    int blocks = (tiles + 31) / 32;                // ~4 tiles per wave to amortize setup
    if (blocks < 1) blocks = 1;
    if (blocks > 4096) blocks = 4096;
    qsa_wmma_kernel<<<blocks, 256, 0, stream>>>(features, out, rows);
}